// YOLOv3Loss_23897198035702
// MI455X (gfx1250) — compile-verified
//
#include <hip/hip_runtime.h>
#include <hip/hip_bf16.h>

#define NCLS 80

typedef __attribute__((ext_vector_type(2))) float v2f;
typedef __attribute__((ext_vector_type(8))) float v8f;

__device__ __constant__ float c_anchW[12] = {8,11,16,23,32,45,64,90,128,180,256,512};
__device__ __constant__ float c_anchH[12] = {24,34,48,68,96,135,192,271,384,540,640,640};

// workspace layout, offsets in 4-byte words (int slots via I, float slots via F)
#define OFF_IMG   0
#define OFF_CLS   128
#define OFF_LAYER 256
#define OFF_A     384
#define OFF_CX    512
#define OFF_CY    640
#define OFF_KEYS  768   // (img,cy,cx) key for "same" matrix
#define OFF_KEYC  896   // (img,a,cy,cx) key for "pos" scatter
#define OFF_TX    1024
#define OFF_TY    1152
#define OFF_TWL   1280
#define OFF_THL   1408
#define OFF_S     1536
#define OFF_BOX   1664  // 128*4 xyxy (normalized)
#define OFF_AW    2176
#define OFF_AH    2304
#define OFF_CNT   2432  // 3 per-layer target counts
#define OFF_ACC   2440  // 3 layers * 16 accumulator slots
#define OFF_TC    2496  // 128*80 target-class matrix

__device__ inline float softplusf(float z) {
  return fmaxf(z, 0.f) + log1pf(expf(-fabsf(z)));
}

// ---------------- Kernel 1: target prep + accumulator zeroing ----------------
__global__ void k_prep(const float* __restrict__ tgt, int N,
                       float* __restrict__ F, int* __restrict__ I) {
  int i = threadIdx.x;                  // 128 threads
  __shared__ float scnt[3];
  if (i < 3) scnt[i] = 0.f;
  if (i < 48) F[OFF_ACC + i] = 0.f;     // zero all per-layer accumulators
  __syncthreads();
  if (i < N) {
    const float* t = tgt + i * 6;
    int img = (int)t[0], cls = (int)t[1];
    float tx = t[2], ty = t[3], tw = t[4], th = t[5];
    // argmax IoU over 12 normalized anchors (first-max like jnp.argmax)
    float best = -1.f; int g = 0;
    for (int k = 0; k < 12; ++k) {
      float aw = c_anchW[k] * (1.f / 416.f), ah = c_anchH[k] * (1.f / 416.f);
      float inter = fminf(aw, tw) * fminf(ah, th);
      float iou = inter / (aw * ah + tw * th - inter + 1e-16f);
      if (iou > best) { best = iou; g = k; }
    }
    int a = g & 3, layer = 2 - (g >> 2);
    float gs = (layer == 0) ? 13.f : ((layer == 1) ? 26.f : 52.f);
    float bx = tx * gs, by = ty * gs;
    float cx = floorf(bx), cy = floorf(by);
    int cxi = (int)cx, cyi = (int)cy;
    I[OFF_IMG + i] = img;  I[OFF_CLS + i] = cls;
    I[OFF_LAYER + i] = layer; I[OFF_A + i] = a;
    I[OFF_CX + i] = cxi;  I[OFF_CY + i] = cyi;
    I[OFF_KEYS + i] = (img * 64 + cyi) * 64 + cxi;
    I[OFF_KEYC + i] = ((img * 4 + a) * 64 + cyi) * 64 + cxi;
    F[OFF_TX + i] = bx - cx;  F[OFF_TY + i] = by - cy;
    F[OFF_TWL + i] = logf(tw * 416.f / c_anchW[g]);
    F[OFF_THL + i] = logf(th * 416.f / c_anchH[g]);
    F[OFF_S + i] = sqrtf(2.f - tw * th);
    F[OFF_BOX + 4 * i + 0] = tx - tw * 0.5f;
    F[OFF_BOX + 4 * i + 1] = ty - th * 0.5f;
    F[OFF_BOX + 4 * i + 2] = tx + tw * 0.5f;
    F[OFF_BOX + 4 * i + 3] = ty + th * 0.5f;
    F[OFF_AW + i] = c_anchW[g];  F[OFF_AH + i] = c_anchH[g];
    atomicAdd(&scnt[layer], 1.f);
  } else if (i < 128) {                 // pad for the WMMA K-loop
    I[OFF_KEYS + i] = -2 - i;           // unique key: matches nothing
    I[OFF_CLS + i]  = -1;               // matches no class
  }
  __syncthreads();
  if (i < 3) F[OFF_CNT + i] = scnt[i];
}

// ---------------- Kernel 2: tc = (same @ onehot) > 0 via V_WMMA_F32_16X16X4_F32
// 8 waves (256 threads): wave w owns row tile M=16w..16w+15.
// Register-blocked: the K loop is outermost and the 5 column-tile accumulators
// (5 x v8f = 40 VGPRs) stay live, so each A-fragment is built once from LDS
// and consumed by 5 back-to-back WMMAs -> no loop-invariant hoisting, no
// scratch spills in the matrix loop (previous version spilled A-fragments).
__global__ void k_tc(const int* __restrict__ I, float* __restrict__ F) {
  __shared__ int skey[128], scls[128];
  int t = threadIdx.x;
  if (t < 128) { skey[t] = I[OFF_KEYS + t]; scls[t] = I[OFF_CLS + t]; }
  __syncthreads();
  int wave = t >> 5, lane = t & 31, lr = lane & 15, hi = lane >> 4;
  int mrow = (wave << 4) + lr;          // A: lane holds row M, lanes>=16 shift K by 2
  int myk = skey[mrow];

  v8f acc[5];
#pragma unroll
  for (int nt = 0; nt < 5; ++nt) acc[nt] = (v8f){0.f,0.f,0.f,0.f,0.f,0.f,0.f,0.f};

  for (int kk = 0; kk < 128; kk += 4) {
    int k0 = kk + hi * 2;
    int key0 = skey[k0], key1 = skey[k0 + 1];
    int cls0 = scls[k0], cls1 = scls[k0 + 1];
    v2f a;
    a.x = (myk == key0) ? 1.f : 0.f;    // same[M][k0]
    a.y = (myk == key1) ? 1.f : 0.f;    // same[M][k0+1]
#pragma unroll
    for (int nt = 0; nt < 5; ++nt) {
      int ncol = (nt << 4) + lr;        // B/D: lane%16 is column N
      v2f b;
      b.x = (cls0 == ncol) ? 1.f : 0.f; // onehot[k0][N]
      b.y = (cls1 == ncol) ? 1.f : 0.f; // onehot[k0+1][N]
      acc[nt] = __builtin_amdgcn_wmma_f32_16x16x4_f32(
          false, a, false, b, (short)0, acc[nt], false, false);
    }
  }
#pragma unroll
  for (int nt = 0; nt < 5; ++nt) {
#pragma unroll
    for (int v = 0; v < 8; ++v) {       // D: VGPR v -> row v (+8 for lanes>=16)
      int row = (wave << 4) + v + (hi << 3);
      F[OFF_TC + row * NCLS + (nt << 4) + lr] = (acc[nt][v] > 0.f) ? 1.f : 0.f;
    }
  }
}

// ---------------- Kernel 3: per-target losses/metrics (one 128-thread block)
__global__ void k_target(const float* __restrict__ x0, const float* __restrict__ x1,
                         const float* __restrict__ x2, int N,
                         float* __restrict__ F, const int* __restrict__ I) {
  __shared__ float sacc[48];            // 3 layers x 16 slots (use 0..9)
  int i = threadIdx.x;
  if (i < 48) sacc[i] = 0.f;
  __syncthreads();
  if (i < N) {
    int l = I[OFF_LAYER + i];
    const float* X = (l == 0) ? x0 : ((l == 1) ? x1 : x2);
    int g = (l == 0) ? 13 : ((l == 1) ? 26 : 52);
    int n = I[OFF_IMG + i], a = I[OFF_A + i];
    int yy = I[OFF_CY + i], xx = I[OFF_CX + i];
    size_t cs = (size_t)g * g;
    const float* base = X + ((size_t)n * 340 + (size_t)a * 85) * cs + (size_t)yy * g + xx;
    float c0 = base[0], c1 = base[cs], c2 = base[2 * cs], c3 = base[3 * cs];
    float c4 = base[4 * cs], c5 = base[5 * cs];
    float p4 = 1.f / (1.f + expf(c4 - c5));       // softmax(x4,x5)[1]
    float s = F[OFF_S + i];
    float tv[4] = {F[OFF_TX + i], F[OFF_TY + i], F[OFF_TWL + i], F[OFF_THL + i]};
    float pv[4] = {c0, c1, c2, c3};
    float lx = 0.f;
    for (int c = 0; c < 4; ++c) {
      float d = fabsf(s * pv[c] - s * tv[c]);
      lx += (d < 1.f) ? (0.5f * d * d) : (d - 0.5f);
    }
    float lo = softplusf(p4) - p4;
    int cls = I[OFF_CLS + i];
    float lc = 0.f, best = 0.f, acat = 0.f; int bi = 0;
    for (int c = 0; c < NCLS; ++c) {
      float pcv = (c == 0) ? 0.f : base[(size_t)(5 + c) * cs];  // pc[0] forced 0
      lc += softplusf(pcv) - pcv * F[OFF_TC + i * NCLS + c];
      if (pcv > best) { best = pcv; bi = c; }
      if (c == cls) acat = pcv;
    }
    float cacc = (cls == bi) ? 1.f : 0.f;
    float tp = (p4 > 0.5f) ? 1.f : 0.f;
    // decoded box (pixel units) vs normalized target box, exactly as reference
    float aw = F[OFF_AW + i], ah = F[OFF_AH + i];
    float spix = 416.f / (float)g;
    float bx = c0 * aw + xx * spix, by = c1 * ah + yy * spix;
    float bw = expf(c2) * aw, bh = expf(c3) * ah;
    float px0 = bx - 0.5f * bw, py0 = by - 0.5f * bh;
    float px1 = bx + 0.5f * bw, py1 = by + 0.5f * bh;
    float a0 = F[OFF_BOX + 4 * i],     a1 = F[OFF_BOX + 4 * i + 1];
    float a2 = F[OFF_BOX + 4 * i + 2], a3 = F[OFF_BOX + 4 * i + 3];
    float iw = fmaxf(fminf(a2, px1) - fmaxf(a0, px0), 0.f);
    float ih = fmaxf(fminf(a3, py1) - fmaxf(a1, py0), 0.f);
    float inter = iw * ih;
    float iou = inter / ((a2 - a0) * (a3 - a1) + (px1 - px0) * (py1 - py0) - inter + 1e-16f);
    float* A = &sacc[l * 16];
    atomicAdd(&A[0], lx);   atomicAdd(&A[1], lo);  atomicAdd(&A[2], lc);
    atomicAdd(&A[3], cacc); atomicAdd(&A[4], p4);  atomicAdd(&A[5], tp);
    atomicAdd(&A[6], (iou > 0.5f)  ? 1.f : 0.f);
    atomicAdd(&A[7], (iou > 0.75f) ? 1.f : 0.f);
    atomicAdd(&A[8], iou);  atomicAdd(&A[9], acat);
  }
  __syncthreads();
  if (i < 48 && (i & 15) < 10) F[OFF_ACC + i] = sacc[i];
}

// ---------------- Kernel 4: per-cell background term (one thread per cell)
__global__ void k_cells(const float* __restrict__ X, int l, int g, int B, int N,
                        float* __restrict__ F, const int* __restrict__ I) {
  __shared__ float tb0[128], tb1[128], tb2[128], tb3[128], tarea[128];
  __shared__ int tkey[128], tlay[128];
  __shared__ float cntl;
  __shared__ float bacc[4];
  int t = threadIdx.x;
  if (t < N) {
    float b0 = F[OFF_BOX + 4 * t],     b1 = F[OFF_BOX + 4 * t + 1];
    float b2 = F[OFF_BOX + 4 * t + 2], b3 = F[OFF_BOX + 4 * t + 3];
    tb0[t] = b0; tb1[t] = b1; tb2[t] = b2; tb3[t] = b3;
    tarea[t] = (b2 - b0) * (b3 - b1);
    tkey[t] = I[OFF_KEYC + t]; tlay[t] = I[OFF_LAYER + t];
  }
  if (t == 0) cntl = F[OFF_CNT + l];
  if (t < 4) bacc[t] = 0.f;
  __syncthreads();
  int id = blockIdx.x * blockDim.x + t;
  int cells = B * 4 * g * g;
  if (id < cells) {
    int xx = id % g;
    int yy = (id / g) % g;
    int a  = (id / (g * g)) & 3;
    int n  = id / (4 * g * g);
    size_t cs = (size_t)g * g;
    const float* base = X + ((size_t)n * 340 + (size_t)a * 85) * cs + (size_t)yy * g + xx;
    float c0 = base[0], c1 = base[cs], c2 = base[2 * cs], c3 = base[3 * cs];
    float c4 = base[4 * cs], c5 = base[5 * cs];
    float p4 = 1.f / (1.f + expf(c4 - c5));
    int aidx = (2 - l) * 4 + a;
    float aw = c_anchW[aidx], ah = c_anchH[aidx];
    float spix = 416.f / (float)g;
    float bx = c0 * aw + xx * spix, by = c1 * ah + yy * spix;
    float bw = expf(c2) * aw, bh = expf(c3) * ah;
    float px0 = bx - 0.5f * bw, py0 = by - 0.5f * bh;
    float px1 = bx + 0.5f * bw, py1 = by + 0.5f * bh;
    float parea = (px1 - px0) * (py1 - py0);
    int mykey = ((n * 4 + a) * 64 + yy) * 64 + xx;
    float vmax = 0.f; int pos = 0;
    for (int i = 0; i < N; ++i) {
      float iw = fminf(tb2[i], px1) - fmaxf(tb0[i], px0); iw = fmaxf(iw, 0.f);
      float ih = fminf(tb3[i], py1) - fmaxf(tb1[i], py0); ih = fmaxf(ih, 0.f);
      float inter = iw * ih;
      float v = inter / (tarea[i] + parea - inter + 1e-16f);
      vmax = fmaxf(vmax, v);
      pos |= (tlay[i] == l) & (tkey[i] == mykey);
    }
    int ign = (vmax > 0.5f) && (cntl > 0.f);
    float bkg = (pos || ign) ? 0.f : 1.f;
    atomicAdd(&bacc[0], bkg);
    atomicAdd(&bacc[1], softplusf(p4) * bkg);
    atomicAdd(&bacc[2], p4 * bkg);
    atomicAdd(&bacc[3], (p4 > 0.5f) ? 1.f : 0.f);
  }
  __syncthreads();
  if (t < 4) atomicAdd(&F[OFF_ACC + l * 16 + 10 + t], bacc[t]);
}

// ---------------- Kernel 5: finalize loss + 3x12 metrics ----------------
__global__ void k_final(const float* __restrict__ F, float* __restrict__ out) {
  if (threadIdx.x == 0 && blockIdx.x == 0) {
    float loss = 0.f;
    for (int l = 0; l < 3; ++l) {
      const float* A = &F[OFF_ACC + l * 16];
      float d = fmaxf(F[OFF_CNT + l], 1.f);
      float lxywh = A[0] / (d * 4.f);
      float lo = A[1] / d;
      float lc = A[2] / (d * (float)NCLS);
      float nb = fmaxf(A[10], 1.f);
      float lb = A[11] / nb;
      loss += lxywh + lo + lc + lb;
      float* m = out + 1 + l * 12;
      m[0] = A[3] / d;               // cacc
      m[1] = A[4] / d;               // conf
      m[2] = A[12] / nb;             // bkgc
      m[3] = A[5] / (A[13] + 1e-6f); // prec = tp/(pp+1e-6)
      m[4] = A[6] / d;               // rc50
      m[5] = A[7] / d;               // rc75
      m[6] = A[8] / d;               // aiou
      m[7] = A[9] / d;               // acat
      m[8] = lxywh; m[9] = lo; m[10] = lc; m[11] = lb;
    }
    out[0] = loss;
  }
}

extern "C" void kernel_launch(void* const* d_in, const int* in_sizes, int n_in,
                              void* d_out, int out_size, void* d_ws, size_t ws_size,
                              hipStream_t stream) {
  const float* x0 = (const float*)d_in[0];
  const float* x1 = (const float*)d_in[1];
  const float* x2 = (const float*)d_in[2];
  const float* tgt = (const float*)d_in[3];
  int B = in_sizes[0] / (340 * 13 * 13);
  int N = in_sizes[3] / 6; if (N > 128) N = 128;
  float* F = (float*)d_ws;
  int*   I = (int*)d_ws;

  k_prep<<<1, 128, 0, stream>>>(tgt, N, F, I);
  k_tc<<<1, 256, 0, stream>>>(I, F);
  k_target<<<1, 128, 0, stream>>>(x0, x1, x2, N, F, I);
  const float* xs[3] = {x0, x1, x2};
  const int gs[3] = {13, 26, 52};
  for (int l = 0; l < 3; ++l) {
    int cells = B * 4 * gs[l] * gs[l];
    int blocks = (cells + 255) / 256;
    k_cells<<<blocks, 256, 0, stream>>>(xs[l], l, gs[l], B, N, F, I);
  }
  k_final<<<1, 1, 0, stream>>>(F, (float*)d_out);
}